// PKGEncoder_72756745994394
// MI455X (gfx1250) — compile-verified
//
#include <hip/hip_runtime.h>

// ---------------------------------------------------------------------------
// PKG R-GCN encoder for MI455X (gfx1250, wave32).
// Aggregate-per-(dst,relation) first (linearity), then dense fp32 WMMA GEMMs
// (V_WMMA_F32_16X16X4_F32).  Bandwidth-bound (~1.7 GB @ 23.3 TB/s), so fp32
// matrix path keeps reference numerics while easily hitting the memory floor.
// A/B operands staged via CDNA5 async global->LDS copies (ASYNCcnt).
// ---------------------------------------------------------------------------

#define Hdim 128
#define WAVES 8
#define ROWS_PER_BLOCK (WAVES * 16)
#define LDS_BYTES (2 * 64 * 1024)  // 64KB W + 8 * 8KB X strips

typedef __attribute__((ext_vector_type(2))) float v2f;
typedef __attribute__((ext_vector_type(8))) float v8f;

// Device-scope f32 atomic add (CDNA5 global_atomic_add_f32, no CAS loop).
__device__ __forceinline__ void gatomic_add(float* p, float v) {
  asm volatile("global_atomic_add_f32 %0, %1, off scope:SCOPE_DEV"
               :: "v"(p), "v"(v) : "memory");
}

// Async global->LDS 16-byte copy (per-lane addresses, tracked by ASYNCcnt).
__device__ __forceinline__ void async_g2l_b128(unsigned lds_off, const void* g) {
  asm volatile("global_load_async_to_lds_b128 %0, %1, off"
               :: "v"(lds_off), "v"(g) : "memory");
}
__device__ __forceinline__ void wait_async0() {
  asm volatile("s_wait_asynccnt 0" ::: "memory");
}

// LDS byte offset of a generic pointer to __shared__ (low 32 bits of the
// LDS-aperture generic address are the LDS offset).
__device__ __forceinline__ unsigned lds_off(const void* p) {
  return (unsigned)(unsigned long long)p;
}

// ---------------------------------------------------------------------------
// Y[M,128] = epilogue( X[M,128] @ W[128,128]^T )  with optional bias,
// optional additive term (scaled by 1/max(deg,1)), optional relu.
// 8 waves/block, wave owns a 16x128 strip = 8 accumulator tiles of
// V_WMMA_F32_16X16X4_F32.  W and the X strip live in LDS with column swizzle
// (k + 4*row) & 127 -> fragment fetches cover all 64 banks exactly once.
// Swizzle is invariant under row+=32 and XOR-64 under row+=16, so two base
// pointers + immediate ds offsets (0/16K/32K/48K) address all 8 tiles.
// M must be a multiple of 128.
// ---------------------------------------------------------------------------
template <int RELU, int HAS_BIAS, int HAS_ADD, int HAS_DEG>
__global__ __launch_bounds__(256, 1) void gemm_xwT(
    const float* __restrict__ X, const float* __restrict__ W,
    const float* __restrict__ bias, const float* __restrict__ addbuf,
    const float* __restrict__ deg, float* __restrict__ Y) {
  extern __shared__ float smem[];  // [0,16384): W ; [16384 + wave*2048): X strip

  const int tid = threadIdx.x;
  const int wave = tid >> 5;
  const int lane = tid & 31;
  const int m0 = (blockIdx.x * WAVES + wave) * 16;
  const int n0 = lane & 15;   // A-row / B-col index within tile
  const int khalf = lane >> 4;

  // --- async stage of W (whole workgroup: 4096 float4, 16 per thread) ------
#pragma unroll
  for (int it = 0; it < 16; ++it) {
    const int i = it * 256 + tid;
    const int row = i >> 5;
    const int c4 = (i & 31) << 2;
    const unsigned lo =
        lds_off(&smem[row * Hdim + ((c4 + 4 * row) & (Hdim - 1))]);
    async_g2l_b128(lo, W + (size_t)row * Hdim + c4);
  }
  // --- async stage of this wave's 16-row X strip (512 float4, 16 per lane) -
  float* xs = smem + 16384 + wave * 2048;
  const float* xsrc = X + (size_t)m0 * Hdim;
#pragma unroll
  for (int j = 0; j < 16; ++j) {
    const int f = j * 32 + lane;
    const int row = f >> 5;
    const int c4 = (f & 31) << 2;
    const unsigned lo =
        lds_off(&xs[row * Hdim + ((c4 + 4 * row) & (Hdim - 1))]);
    async_g2l_b128(lo, xsrc + (size_t)row * Hdim + c4);
  }
  wait_async0();     // our own async writes visible
  __syncthreads();   // everyone's W slice visible

  v8f acc[8];
#pragma unroll
  for (int t = 0; t < 8; ++t) {
    float init = 0.0f;
    if (HAS_BIAS) init = bias[t * 16 + n0];
#pragma unroll
    for (int v = 0; v < 8; ++v) acc[t][v] = init;
  }

  if (HAS_ADD) __builtin_prefetch(addbuf + (size_t)(m0 + n0) * Hdim, 0, 3);

  // K = 128 in 32 steps of 4.  A frag: {x[m,k],x[m,k+1]} (m=n0, k-pair by
  // lane half); B frag: {W[n,k],W[n,k+1]} with n = t*16 + n0 (B = W^T tile).
#pragma unroll 4
  for (int kb = 0; kb < 32; ++kb) {
    const int k = kb * 4 + khalf * 2;
    const int s0 = (k + 4 * n0) & (Hdim - 1);
    const float* pa = &xs[n0 * Hdim + s0];
    const float* pb0 = &smem[n0 * Hdim + s0];               // tiles 0,2,4,6
    const float* pb1 = &smem[(n0 + 16) * Hdim + (s0 ^ 64)]; // tiles 1,3,5,7
    const v2f a = *(const v2f*)pa;
#pragma unroll
    for (int u = 0; u < 4; ++u) {
      const v2f bE = *(const v2f*)(pb0 + u * 4096);
      const v2f bO = *(const v2f*)(pb1 + u * 4096);
      acc[2 * u] = __builtin_amdgcn_wmma_f32_16x16x4_f32(
          false, a, false, bE, (short)0, acc[2 * u], false, false);
      acc[2 * u + 1] = __builtin_amdgcn_wmma_f32_16x16x4_f32(
          false, a, false, bO, (short)0, acc[2 * u + 1], false, false);
    }
  }

  // C/D layout: vgpr v -> M = m0 + v + 8*(lane>=16), N = t*16 + lane%16.
#pragma unroll
  for (int t = 0; t < 8; ++t) {
    const int n = t * 16 + n0;
#pragma unroll
    for (int v = 0; v < 8; ++v) {
      const int m = m0 + v + (khalf << 3);
      float val = acc[t][v];
      if (HAS_ADD) {
        float addv = addbuf[(size_t)m * Hdim + n];
        if (HAS_DEG) addv *= __builtin_amdgcn_rcpf(fmaxf(deg[m], 1.0f));
        val += addv;
      }
      if (RELU) val = fmaxf(val, 0.0f);
      Y[(size_t)m * Hdim + n] = val;
    }
  }
}

// ---------------------------------------------------------------------------
// Per-relation scatter: for edges with type==r, bin[dst] += x[src].
// One wave per edge, float4 per lane (32 lanes * 4 = 128 = H).
// ---------------------------------------------------------------------------
__global__ __launch_bounds__(256) void scatter_rel(
    const float* __restrict__ X, const int* __restrict__ src,
    const int* __restrict__ dst, const int* __restrict__ etype,
    float* __restrict__ bin, int E, int r) {
  const int gid = blockIdx.x * blockDim.x + threadIdx.x;
  const int e = gid >> 5;
  const int lane = gid & 31;
  if (e >= E) return;
  if (etype[e] != r) return;
  const int s = src[e];
  const int d = dst[e];
  const float4 v = *(const float4*)(X + (size_t)s * Hdim + lane * 4);
  float* p = bin + (size_t)d * Hdim + lane * 4;
  gatomic_add(p + 0, v.x);
  gatomic_add(p + 1, v.y);
  gatomic_add(p + 2, v.z);
  gatomic_add(p + 3, v.w);
}

__global__ __launch_bounds__(256) void deg_kernel(const int* __restrict__ dst,
                                                  float* __restrict__ deg, int E) {
  const int e = blockIdx.x * blockDim.x + threadIdx.x;
  if (e < E) gatomic_add(deg + dst[e], 1.0f);
}

__global__ void embed_kernel(const int* __restrict__ cid, const int* __restrict__ kid,
                             const float* __restrict__ cemb,
                             const float* __restrict__ kemb,
                             float* __restrict__ xo) {
  const int node = blockIdx.x;
  const int h = threadIdx.x;
  const int c = cid[node];
  const int k = kid[node];
  xo[(size_t)node * Hdim + h] =
      cemb[(size_t)c * Hdim + h] + kemb[(size_t)k * Hdim + h];
}

__global__ void mask_kernel(const float* __restrict__ x, const int* __restrict__ mask,
                            float* __restrict__ out) {
  const int node = blockIdx.x;
  const int h = threadIdx.x;
  out[(size_t)node * Hdim + h] = x[(size_t)node * Hdim + h] * (float)mask[node];
}

__global__ void zero_kernel(float4* __restrict__ p, int n4) {
  const int i = blockIdx.x * blockDim.x + threadIdx.x;
  if (i < n4) p[i] = make_float4(0.f, 0.f, 0.f, 0.f);
}

// ---------------------------------------------------------------------------
extern "C" void kernel_launch(void* const* d_in, const int* in_sizes, int n_in,
                              void* d_out, int out_size, void* d_ws, size_t ws_size,
                              hipStream_t stream) {
  (void)n_in; (void)out_size; (void)ws_size;
  const int* cid = (const int*)d_in[0];
  const int* kid = (const int*)d_in[1];
  const int* nmask = (const int*)d_in[2];
  const int* eidx = (const int*)d_in[3];
  const int* etype = (const int*)d_in[4];
  const float* cemb = (const float*)d_in[5];
  const float* kemb = (const float*)d_in[6];
  const float* projW = (const float*)d_in[7];
  const float* projB = (const float*)d_in[8];
  const float* selfW = (const float*)d_in[9];
  const float* selfB = (const float*)d_in[10];
  const float* relW = (const float*)d_in[11];

  const int M = in_sizes[0];  // 32768 packed nodes (multiple of 128)
  const int E = in_sizes[4];  // 524288 edges
  const int* esrc = eidx;
  const int* edst = eidx + E;

  // Workspace carve (~67 MB): xA, xB, bin, agg [M,128] f32 each + deg [M].
  float* xA = (float*)d_ws;
  float* xB = xA + (size_t)M * Hdim;
  float* bin = xB + (size_t)M * Hdim;
  float* agg = bin + (size_t)M * Hdim;
  float* deg = agg + (size_t)M * Hdim;

  const int gemmGrid = M / ROWS_PER_BLOCK;
  const int nf4 = M * Hdim / 4;

  // Degrees (count of incoming edges per node).
  zero_kernel<<<(M / 4 + 255) / 256, 256, 0, stream>>>((float4*)deg, M / 4);
  deg_kernel<<<(E + 255) / 256, 256, 0, stream>>>(edst, deg, E);

  // Input embedding + projection: xA = (cemb[cid]+kemb[kid]) @ projW^T + b.
  embed_kernel<<<M, Hdim, 0, stream>>>(cid, kid, cemb, kemb, xB);
  gemm_xwT<0, 1, 0, 0><<<gemmGrid, 256, LDS_BYTES, stream>>>(
      xB, projW, projB, nullptr, nullptr, xA);

  float* cur = xA;
  float* nxt = xB;
  for (int l = 0; l < 2; ++l) {
    zero_kernel<<<(nf4 + 255) / 256, 256, 0, stream>>>((float4*)agg, nf4);
    for (int r = 0; r < 8; ++r) {
      // bin = segment_sum over edges of type r of cur[src], keyed by dst.
      zero_kernel<<<(nf4 + 255) / 256, 256, 0, stream>>>((float4*)bin, nf4);
      scatter_rel<<<(E * 32 + 255) / 256, 256, 0, stream>>>(cur, esrc, edst,
                                                            etype, bin, E, r);
      // agg += bin @ rel_W[l,r]^T   (linearity of the relation transform)
      gemm_xwT<0, 0, 1, 0><<<gemmGrid, 256, LDS_BYTES, stream>>>(
          bin, relW + (size_t)(l * 8 + r) * Hdim * Hdim, nullptr, agg,
          nullptr, agg);
    }
    // nxt = relu(cur @ self_W[l]^T + self_b[l] + agg / max(deg,1))
    gemm_xwT<1, 1, 1, 1><<<gemmGrid, 256, LDS_BYTES, stream>>>(
        cur, selfW + (size_t)l * Hdim * Hdim, selfB + (size_t)l * Hdim, agg,
        deg, nxt);
    float* t = cur; cur = nxt; nxt = t;
  }

  mask_kernel<<<M, Hdim, 0, stream>>>(cur, nmask, (float*)d_out);
}